// GraphEncoder_31344671326628
// MI455X (gfx1250) — compile-verified
//
#include <hip/hip_runtime.h>
#include <hip/hip_bf16.h>
#include <cfloat>

typedef float v2f __attribute__((ext_vector_type(2)));
typedef float v8f __attribute__((ext_vector_type(8)));

#define DIM    128   // H*hid
#define HID    64
#define NHEAD  2
#define NGRAPH 512

// ---------- order-preserving float<->uint encoding for atomic max ----------
__device__ __forceinline__ unsigned fenc(float f) {
  unsigned u = __float_as_uint(f);
  return (u & 0x80000000u) ? ~u : (u | 0x80000000u);
}
__device__ __forceinline__ float fdec(unsigned u) {
  return (u & 0x80000000u) ? __uint_as_float(u & 0x7FFFFFFFu)
                           : __uint_as_float(~u);
}
// host-side constant: fenc(-FLT_MAX) = ~0xFF7FFFFF = 0x00800000
#define ENC_NEG_MAX 0x00800000u

// ------------------------------ utilities ---------------------------------
__global__ void k_zero_f32(float* __restrict__ p, int n) {
  int i = blockIdx.x * blockDim.x + threadIdx.x;
  if (i < n) p[i] = 0.0f;
}
__global__ void k_init_u32(unsigned* __restrict__ p, unsigned v, int n) {
  int i = blockIdx.x * blockDim.x + threadIdx.x;
  if (i < n) p[i] = v;
}

// ------------------------- embedding gather -------------------------------
__global__ void k_embed(const int* __restrict__ x, const float* __restrict__ emb,
                        float* __restrict__ xe, int N) {
  int i = blockIdx.x * blockDim.x + threadIdx.x;
  if (i >= N * DIM) return;
  int n = i >> 7, f = i & (DIM - 1);
  int tok = x[n];
  xe[i] = (tok == 0) ? 0.0f : emb[tok * DIM + f];   // emb.at[0].set(0)
}

// ---------------- fp32 WMMA GEMM: C[N x 128] = A[N x 128] * W[128 x 128] ---
// 256 threads = 8 waves; block handles a 16-row strip, wave w -> cols [16w,16w+16)
// V_WMMA_F32_16X16X4_F32: A 16x4 (2 VGPR/lane), B 4x16 (2 VGPR/lane), C 16x16 (8 VGPR/lane)
__global__ void __launch_bounds__(256)
k_gemm128_wmma(const float* __restrict__ A, const float* __restrict__ W,
               float* __restrict__ C, int N) {
  __shared__ float As[16][DIM + 4];          // +4 pad: kills LDS bank conflicts
  const int row0 = blockIdx.x * 16;
  const int tid  = threadIdx.x;

  for (int i = tid; i < 16 * DIM; i += 256) {
    int r = i >> 7, c = i & (DIM - 1);
    int gr = row0 + r;
    As[r][c] = (gr < N) ? A[(size_t)gr * DIM + c] : 0.0f;
  }
  __syncthreads();

  const int wave = tid >> 5;
  const int lane = tid & 31;
  const int n0   = wave * 16;
  const int mr   = lane & 15;          // A row / C column-within-tile
  const int kh   = (lane >> 4) * 2;    // K sub-offset per half-wave
  const int col  = n0 + mr;            // B/C column

  v8f acc = {};
  for (int k0 = 0; k0 < DIM; k0 += 4) {
    v2f a, b;
    a.x = As[mr][k0 + kh];
    a.y = As[mr][k0 + kh + 1];
    b.x = W[(size_t)(k0 + kh) * DIM + col];
    b.y = W[(size_t)(k0 + kh + 1) * DIM + col];
    acc = __builtin_amdgcn_wmma_f32_16x16x4_f32(
        /*neg_a=*/false, a, /*neg_b=*/false, b,
        /*c_mod=*/(short)0, acc, /*reuse_a=*/false, /*reuse_b=*/false);
  }

  const int mbase = (lane >> 4) * 8;   // C layout: vgpr r -> M = r (+8 for hi lanes)
#pragma unroll
  for (int r = 0; r < 8; ++r) {
    int grow = row0 + mbase + r;
    if (grow < N) C[(size_t)grow * DIM + col] = acc[r];
  }
}

// ------------- per-node attention scores: one wave per node ----------------
// lanes 0-15 -> head 0 (f 0..63), lanes 16-31 -> head 1 (f 64..127), 4 f/lane
__global__ void k_scores(const float* __restrict__ h,
                         const float* __restrict__ att_s,
                         const float* __restrict__ att_d,
                         float* __restrict__ a_s, float* __restrict__ a_d, int N) {
  int wid  = (blockIdx.x * blockDim.x + threadIdx.x) >> 5;
  int lane = threadIdx.x & 31;
  if (wid >= N) return;
  float ps = 0.f, pd = 0.f;
  int fbase = lane * 4;
#pragma unroll
  for (int j = 0; j < 4; ++j) {
    int f = fbase + j;
    float v = h[(size_t)wid * DIM + f];
    ps += v * att_s[f];
    pd += v * att_d[f];
  }
  for (int m = 8; m >= 1; m >>= 1) {       // reduce within each 16-lane head group
    ps += __shfl_xor(ps, m, 32);
    pd += __shfl_xor(pd, m, 32);
  }
  if ((lane & 15) == 0) {
    int head = lane >> 4;
    a_s[wid * NHEAD + head] = ps;
    a_d[wid * NHEAD + head] = pd;
  }
}

// --------------------------- edge softmax passes ---------------------------
__global__ void k_edge_alpha_max(const int* __restrict__ ei, int E, int N,
                                 const float* __restrict__ a_s,
                                 const float* __restrict__ a_d,
                                 float* __restrict__ ealpha,
                                 unsigned* __restrict__ menc) {
  int e = blockIdx.x * blockDim.x + threadIdx.x;
  int ET = E + N;
  if (e >= ET) return;
  int src = (e < E) ? ei[e]     : (e - E);
  int dst = (e < E) ? ei[E + e] : (e - E);
#pragma unroll
  for (int hh = 0; hh < NHEAD; ++hh) {
    float al = a_s[src * NHEAD + hh] + a_d[dst * NHEAD + hh];
    al = (al >= 0.f) ? al : 0.2f * al;           // leaky(0.2)
    ealpha[(size_t)e * NHEAD + hh] = al;
    atomicMax(&menc[dst * NHEAD + hh], fenc(al));
  }
}

__global__ void k_edge_exp_sum(const int* __restrict__ ei, int E, int N,
                               const unsigned* __restrict__ menc,
                               float* __restrict__ ealpha,
                               float* __restrict__ s) {
  int e = blockIdx.x * blockDim.x + threadIdx.x;
  int ET = E + N;
  if (e >= ET) return;
  int dst = (e < E) ? ei[E + e] : (e - E);
#pragma unroll
  for (int hh = 0; hh < NHEAD; ++hh) {
    float ev = __expf(ealpha[(size_t)e * NHEAD + hh] - fdec(menc[dst * NHEAD + hh]));
    ealpha[(size_t)e * NHEAD + hh] = ev;
    atomicAdd(&s[dst * NHEAD + hh], ev);
  }
}

// one wave per edge; 4 floats/lane; lane group -> head
__global__ void k_edge_aggregate(const int* __restrict__ ei, int E, int N,
                                 const float* __restrict__ h,
                                 const float* __restrict__ ealpha,
                                 const float* __restrict__ s,
                                 float* __restrict__ accum) {
  int wid  = (blockIdx.x * blockDim.x + threadIdx.x) >> 5;
  int lane = threadIdx.x & 31;
  int ET = E + N;
  if (wid >= ET) return;
  int src = (wid < E) ? ei[wid]     : (wid - E);
  int dst = (wid < E) ? ei[E + wid] : (wid - E);
  int head = lane >> 4;
  float w = ealpha[(size_t)wid * NHEAD + head] / (s[dst * NHEAD + head] + 1e-16f);
  int fbase = lane * 4;
#pragma unroll
  for (int j = 0; j < 4; ++j) {
    int f = fbase + j;
    atomicAdd(&accum[(size_t)dst * DIM + f], h[(size_t)src * DIM + f] * w);
  }
}

__global__ void k_bias_act(float* __restrict__ buf, const float* __restrict__ bias,
                           const float* __restrict__ alpha_p, int N, int do_prelu) {
  int i = blockIdx.x * blockDim.x + threadIdx.x;
  if (i >= N * DIM) return;
  float v = buf[i] + bias[i & (DIM - 1)];
  if (do_prelu) {
    float a = *alpha_p;
    v = (v >= 0.f) ? v : a * v;
  }
  buf[i] = v;
}

// ------------------------ gate MLP: 128 -> 64 -> 1 -------------------------
__global__ void __launch_bounds__(64)
k_gate(const float* __restrict__ out, const float* __restrict__ gW1,
       const float* __restrict__ gb1, const float* __restrict__ ga_p,
       const float* __restrict__ gW2, const float* __restrict__ gb2,
       const int* __restrict__ batch, float* __restrict__ gate,
       unsigned* __restrict__ gmenc, int N) {
  __shared__ float row[DIM];
  __shared__ float part[HID];
  int n = blockIdx.x;
  int t = threadIdx.x;                 // 0..63
  row[t]      = out[(size_t)n * DIM + t];
  row[t + 64] = out[(size_t)n * DIM + t + 64];
  __syncthreads();
  float acc = gb1[t];
  for (int k = 0; k < DIM; ++k) acc += row[k] * gW1[k * HID + t];
  float ga = *ga_p;
  acc = (acc >= 0.f) ? acc : ga * acc;
  part[t] = acc * gW2[t];
  __syncthreads();
  for (int st = 32; st >= 1; st >>= 1) {
    if (t < st) part[t] += part[t + st];
    __syncthreads();
  }
  if (t == 0) {
    float g = part[0] + gb2[0];
    gate[n] = g;
    atomicMax(&gmenc[batch[n]], fenc(g));
  }
}

__global__ void k_gate_exp(const int* __restrict__ batch,
                           const unsigned* __restrict__ gmenc,
                           float* __restrict__ gate, float* __restrict__ gs, int N) {
  int n = blockIdx.x * blockDim.x + threadIdx.x;
  if (n >= N) return;
  int g = batch[n];
  float ge = __expf(gate[n] - fdec(gmenc[g]));
  gate[n] = ge;
  atomicAdd(&gs[g], ge);
}

// one wave per node: pooled hidden accumulation
__global__ void k_hidden(const int* __restrict__ batch,
                         const float* __restrict__ gate, const float* __restrict__ gs,
                         const float* __restrict__ out, float* __restrict__ hidden, int N) {
  int wid  = (blockIdx.x * blockDim.x + threadIdx.x) >> 5;
  int lane = threadIdx.x & 31;
  if (wid >= N) return;
  int g = batch[wid];
  float w = gate[wid] / (gs[g] + 1e-16f);
  int fbase = lane * 4;
#pragma unroll
  for (int j = 0; j < 4; ++j)
    atomicAdd(&hidden[(size_t)g * DIM + fbase + j],
              out[(size_t)wid * DIM + fbase + j] * w);
}

// ------------------------------- launcher ----------------------------------
extern "C" void kernel_launch(void* const* d_in, const int* in_sizes, int n_in,
                              void* d_out, int out_size, void* d_ws, size_t ws_size,
                              hipStream_t stream) {
  const int*   x     = (const int*)d_in[0];
  const int*   ei    = (const int*)d_in[1];
  const int*   batch = (const int*)d_in[2];
  const float* emb   = (const float*)d_in[3];
  const float* W1    = (const float*)d_in[4];
  const float* as1   = (const float*)d_in[5];
  const float* ad1   = (const float*)d_in[6];
  const float* b1    = (const float*)d_in[7];
  const float* a1    = (const float*)d_in[8];
  const float* W2    = (const float*)d_in[9];
  const float* as2   = (const float*)d_in[10];
  const float* ad2   = (const float*)d_in[11];
  const float* b2    = (const float*)d_in[12];
  const float* gW1   = (const float*)d_in[13];
  const float* gb1   = (const float*)d_in[14];
  const float* ga    = (const float*)d_in[15];
  const float* gW2   = (const float*)d_in[16];
  const float* gb2   = (const float*)d_in[17];

  const int N  = in_sizes[0];        // x is (N,1)
  const int E  = in_sizes[1] / 2;
  const int ET = E + N;

  float* out    = (float*)d_out;                 // N*128
  float* hidden = out + (size_t)N * DIM;         // 512*128

  // workspace carve-up (all fits easily in L2: ~60 MB total)
  char* w = (char*)d_ws;
  float*    bufA   = (float*)w;    w += (size_t)N * DIM * 4;
  float*    bufB   = (float*)w;    w += (size_t)N * DIM * 4;
  float*    a_s    = (float*)w;    w += (size_t)N * NHEAD * 4;
  float*    a_d    = (float*)w;    w += (size_t)N * NHEAD * 4;
  unsigned* menc   = (unsigned*)w; w += (size_t)N * NHEAD * 4;
  float*    svec   = (float*)w;    w += (size_t)N * NHEAD * 4;
  float*    ealpha = (float*)w;    w += (size_t)ET * NHEAD * 4;
  float*    gatev  = (float*)w;    w += (size_t)N * 4;
  unsigned* gmenc  = (unsigned*)w; w += NGRAPH * 4;
  float*    gs     = (float*)w;    w += NGRAPH * 4;

  auto cdiv = [](long a, long b) { return (int)((a + b - 1) / b); };
  const dim3 B(256);

  // embedding
  k_embed<<<cdiv((long)N * DIM, 256), B, 0, stream>>>(x, emb, bufA, N);

  // ===================== GAT layer 1 =====================
  k_gemm128_wmma<<<cdiv(N, 16), B, 0, stream>>>(bufA, W1, bufB, N);
  k_scores<<<cdiv((long)N * 32, 256), B, 0, stream>>>(bufB, as1, ad1, a_s, a_d, N);
  k_init_u32<<<cdiv((long)N * NHEAD, 256), B, 0, stream>>>(menc, ENC_NEG_MAX, N * NHEAD);
  k_zero_f32<<<cdiv((long)N * NHEAD, 256), B, 0, stream>>>(svec, N * NHEAD);
  k_zero_f32<<<cdiv((long)N * DIM, 256), B, 0, stream>>>(bufA, N * DIM);
  k_edge_alpha_max<<<cdiv(ET, 256), B, 0, stream>>>(ei, E, N, a_s, a_d, ealpha, menc);
  k_edge_exp_sum<<<cdiv(ET, 256), B, 0, stream>>>(ei, E, N, menc, ealpha, svec);
  k_edge_aggregate<<<cdiv((long)ET * 32, 256), B, 0, stream>>>(ei, E, N, bufB, ealpha, svec, bufA);
  k_bias_act<<<cdiv((long)N * DIM, 256), B, 0, stream>>>(bufA, b1, a1, N, 1);

  // ===================== GAT layer 2 =====================
  k_gemm128_wmma<<<cdiv(N, 16), B, 0, stream>>>(bufA, W2, bufB, N);
  k_scores<<<cdiv((long)N * 32, 256), B, 0, stream>>>(bufB, as2, ad2, a_s, a_d, N);
  k_init_u32<<<cdiv((long)N * NHEAD, 256), B, 0, stream>>>(menc, ENC_NEG_MAX, N * NHEAD);
  k_zero_f32<<<cdiv((long)N * NHEAD, 256), B, 0, stream>>>(svec, N * NHEAD);
  k_zero_f32<<<cdiv((long)N * DIM, 256), B, 0, stream>>>(out, N * DIM);
  k_edge_alpha_max<<<cdiv(ET, 256), B, 0, stream>>>(ei, E, N, a_s, a_d, ealpha, menc);
  k_edge_exp_sum<<<cdiv(ET, 256), B, 0, stream>>>(ei, E, N, menc, ealpha, svec);
  k_edge_aggregate<<<cdiv((long)ET * 32, 256), B, 0, stream>>>(ei, E, N, bufB, ealpha, svec, out);
  k_bias_act<<<cdiv((long)N * DIM, 256), B, 0, stream>>>(out, b2, a1, N, 0);

  // ===================== graph gating + pooling =====================
  k_init_u32<<<cdiv(NGRAPH, 256), B, 0, stream>>>(gmenc, ENC_NEG_MAX, NGRAPH);
  k_zero_f32<<<cdiv(NGRAPH, 256), B, 0, stream>>>(gs, NGRAPH);
  k_zero_f32<<<cdiv((long)NGRAPH * DIM, 256), B, 0, stream>>>(hidden, NGRAPH * DIM);
  k_gate<<<N, 64, 0, stream>>>(out, gW1, gb1, ga, gW2, gb2, batch, gatev, gmenc, N);
  k_gate_exp<<<cdiv(N, 256), B, 0, stream>>>(batch, gmenc, gatev, gs, N);
  k_hidden<<<cdiv((long)N * 32, 256), B, 0, stream>>>(batch, gatev, gs, out, hidden, N);
}